// CenterHead_27882927686223
// MI455X (gfx1250) — compile-verified
//
#include <hip/hip_runtime.h>
#include <hip/hip_bf16.h>
#include <stdint.h>

// Problem constants (B=1, C=80, H=W=512, K=100)
#define C_      80
#define H_      512
#define W_      512
#define K_      100
#define NPLANE  (H_ * W_)
#define TX      128               // tile width (interior)
#define TY      16                // tile height (interior)
#define LDSW    (TX + 4)          // 132: +2 halo, +2 bank pad
#define LDSH    (TY + 2)          // 18
#define NBINS   4096
#define CAPMAX  32768
// fkey(3.5f) >> 20 : 3.5f = 0x40600000 -> key 0xC0600000 -> bin 0xC06.
// Peaks with logit >= 3.5 are optimistically collected in pass 1.
#define STATIC_BIN 0xC06u

// ---- CDNA5 async global->LDS (probe-verified present on this toolchain) ----
#if defined(__has_builtin)
#  if __has_builtin(__builtin_amdgcn_global_load_async_to_lds_b32)
#    define USE_ASYNC_LDS 1
#  endif
#endif

typedef __attribute__((address_space(1))) int* gptr_i32;
typedef __attribute__((address_space(3))) int* lptr_i32;

__device__ __forceinline__ unsigned fkey(float f) {
  // order-preserving flip: larger float -> larger unsigned
  unsigned u = __float_as_uint(f);
  return (u & 0x80000000u) ? ~u : (u | 0x80000000u);
}
__device__ __forceinline__ float unkey(unsigned k) {
  unsigned u = (k & 0x80000000u) ? (k & 0x7FFFFFFFu) : ~k;
  return __uint_as_float(u);
}

// Stage a (TX+2) x (TY+2) halo tile of one class plane into LDS via async
// global->LDS copies. Out-of-plane cells get -inf (reduce_window pads -inf).
static __device__ __forceinline__ void stage_tile(float* sm, const float* plane,
                                                  int tx0, int ty0) {
  const float NEG_INF = -__builtin_inff();
  for (int i = threadIdx.x; i < LDSH * (TX + 2); i += blockDim.x) {
    int lr = i / (TX + 2);
    int lc = i - lr * (TX + 2);
    int gr = ty0 - 1 + lr;
    int gc = tx0 - 1 + lc;
    float* dst = &sm[lr * LDSW + lc];
    bool ok = (gr >= 0) & (gr < H_) & (gc >= 0) & (gc < W_);
    if (!ok) {
      *dst = NEG_INF;
    } else {
#if USE_ASYNC_LDS
      __builtin_amdgcn_global_load_async_to_lds_b32(
          (gptr_i32)(plane + (size_t)gr * W_ + gc),
          (lptr_i32)dst, 0, 0);
#else
      *dst = plane[(size_t)gr * W_ + gc];
#endif
    }
  }
#if USE_ASYNC_LDS
#  if __has_builtin(__builtin_amdgcn_s_wait_asynccnt)
  __builtin_amdgcn_s_wait_asynccnt(0);
#  else
  asm volatile("s_wait_asynccnt 0" ::: "memory");
#  endif
#endif
  __syncthreads();
}

__device__ __forceinline__ float nbr_max(const float* p) {
  return fmaxf(fmaxf(fmaxf(p[-1], p[1]), fmaxf(p[-LDSW], p[LDSW])),
               fmaxf(fmaxf(p[-LDSW - 1], p[-LDSW + 1]),
                     fmaxf(p[LDSW - 1], p[LDSW + 1])));
}

__global__ void zero_ws_kernel(unsigned* __restrict__ hist, unsigned* __restrict__ meta) {
  int i = blockIdx.x * blockDim.x + threadIdx.x;
  if (i < NBINS) hist[i] = 0u;
  if (i < 4) meta[i] = 0u;
}

// Pass 1: NMS on raw logits (monotone-equivalent to sigmoid space).
// Builds exact 4096-bin histogram of peak keys (per-block LDS histogram,
// flushed once -> low global-atomic contention) and optimistically collects
// every peak with bin >= STATIC_BIN into the compact candidate list.
__global__ void __launch_bounds__(256)
nms_hist_kernel(const float* __restrict__ hmap, unsigned* __restrict__ hist,
                unsigned* __restrict__ meta, unsigned* __restrict__ vals,
                unsigned* __restrict__ ids, unsigned cap) {
  __shared__ float sm[LDSH * LDSW];
  __shared__ unsigned lhist[NBINS];
  const int tilesX = W_ / TX;                    // 4
  const int tilesPP = tilesX * (H_ / TY);        // 128 per plane
  int c = blockIdx.x / tilesPP;
  int t = blockIdx.x - c * tilesPP;
  int ty0 = (t / tilesX) * TY;
  int tx0 = (t % tilesX) * TX;
  const float* plane = hmap + (size_t)c * NPLANE;
  __builtin_prefetch(plane + (size_t)ty0 * W_ + tx0, 0, 1);   // global_prefetch_b8
  for (int i = threadIdx.x; i < NBINS; i += blockDim.x) lhist[i] = 0u;
  stage_tile(sm, plane, tx0, ty0);               // barrier also covers lhist zeroing
  for (int i = threadIdx.x; i < TX * TY; i += blockDim.x) {
    int lr = i / TX, lc = i - lr * TX;
    const float* p = &sm[(lr + 1) * LDSW + (lc + 1)];
    float v = p[0];
    if (v >= nbr_max(p)) {                       // hmax == heat  <=>  center >= 8 nbrs
      unsigned key = fkey(v);
      unsigned bin = key >> 20;
      atomicAdd(&lhist[bin], 1u);
      if (bin >= STATIC_BIN) {
        unsigned slot = atomicAdd(&meta[1], 1u);
        if (slot < cap) {
          vals[slot] = key;
          ids[slot] = (unsigned)(c * NPLANE + (ty0 + lr) * W_ + (tx0 + lc)); // < 2^25
        }
      }
    }
  }
  __syncthreads();
  for (int i = threadIdx.x; i < NBINS; i += blockDim.x) {
    unsigned cnt = lhist[i];
    if (cnt) atomicAdd(&hist[i], cnt);
  }
}

// Pass 2: exact coverage check. If the >=STATIC_BIN candidate count covers K
// and fit in the list, the optimistic list provably contains the global top-K
// (bin order == key order) -> fast path. Otherwise compute the exact bin
// threshold T and request a re-collect.
__global__ void thresh_kernel(const unsigned* __restrict__ hist, unsigned* __restrict__ meta,
                              unsigned cap) {
  if (threadIdx.x == 0 && blockIdx.x == 0) {
    unsigned cumS = 0;
    for (int b = NBINS - 1; b >= (int)STATIC_BIN; --b) cumS += hist[b];
    if (cumS >= K_ && cumS <= cap) {
      meta[3] = 0u;                              // fast path: list already valid
    } else {
      meta[3] = 1u;                              // fallback: rebuild list
      meta[1] = 0u;
      unsigned cum = 0;
      int T = 0;
      for (int b = NBINS - 1; b >= 0; --b) {
        cum += hist[b];
        if (cum >= K_) { T = b; break; }
      }
      meta[0] = (unsigned)T;
    }
  }
}

// Pass 3 (fallback only; early-exits on the device flag): re-run NMS and
// compact all peaks with bin >= T.
__global__ void __launch_bounds__(256)
collect_kernel(const float* __restrict__ hmap, unsigned* __restrict__ meta,
               unsigned* __restrict__ vals, unsigned* __restrict__ ids, unsigned cap) {
  if (meta[3] == 0u) return;                     // fast path taken: nothing to do
  __shared__ float sm[LDSH * LDSW];
  const int tilesX = W_ / TX;
  const int tilesPP = tilesX * (H_ / TY);
  int c = blockIdx.x / tilesPP;
  int t = blockIdx.x - c * tilesPP;
  int ty0 = (t / tilesX) * TY;
  int tx0 = (t % tilesX) * TX;
  const float* plane = hmap + (size_t)c * NPLANE;
  unsigned T = meta[0];
  stage_tile(sm, plane, tx0, ty0);
  for (int i = threadIdx.x; i < TX * TY; i += blockDim.x) {
    int lr = i / TX, lc = i - lr * TX;
    const float* p = &sm[(lr + 1) * LDSW + (lc + 1)];
    float v = p[0];
    if (v >= nbr_max(p)) {
      unsigned key = fkey(v);
      if ((key >> 20) >= T) {
        unsigned slot = atomicAdd(&meta[1], 1u);
        if (slot < cap) {
          vals[slot] = key;
          ids[slot] = (unsigned)(c * NPLANE + (ty0 + lr) * W_ + (tx0 + lc));
        }
      }
    }
  }
}

// Pass 4: single-block top-K over the compact list, gather + write [K,7].
__global__ void __launch_bounds__(256)
finalize_kernel(const float* __restrict__ regs, const float* __restrict__ wh,
                const float* __restrict__ rot, const unsigned* __restrict__ meta,
                const unsigned* __restrict__ vals, const unsigned* __restrict__ ids,
                float* __restrict__ out, unsigned cap) {
  __shared__ unsigned long long red[256];
  __shared__ unsigned long long prevBest;
  __shared__ unsigned selKey[K_];
  __shared__ unsigned selId[K_];
  int tid = threadIdx.x;
  unsigned M = meta[1];
  if (M > cap) M = cap;
  if (tid == 0) prevBest = ~0ull;
  __syncthreads();
  for (int k = 0; k < K_; ++k) {
    unsigned long long bound = prevBest;
    unsigned long long best = 0ull;
    for (unsigned i = tid; i < M; i += blockDim.x) {
      // key desc, then (class,pos) asc — matches lax.top_k stability for ties
      unsigned long long cand = ((unsigned long long)vals[i] << 25) |
                                (unsigned long long)(0x1FFFFFFu - (ids[i] & 0x1FFFFFFu));
      if (cand < bound && cand > best) best = cand;
    }
    red[tid] = best;
    __syncthreads();
    for (int s = 128; s > 0; s >>= 1) {
      if (tid < s && red[tid + s] > red[tid]) red[tid] = red[tid + s];
      __syncthreads();
    }
    if (tid == 0) {
      unsigned long long b = red[0];
      prevBest = b;
      if (b == 0ull) {
        selKey[k] = 0u; selId[k] = 0xFFFFFFFFu;
      } else {
        selKey[k] = (unsigned)(b >> 25);
        selId[k] = 0x1FFFFFFu - (unsigned)(b & 0x1FFFFFFu);
      }
    }
    __syncthreads();
  }
  if (tid < K_) {
    float* o = out + tid * 7;
    unsigned id = selId[tid];
    if (id == 0xFFFFFFFFu) {
      for (int j = 0; j < 7; ++j) o[j] = 0.0f;
    } else {
      int c = (int)(id / NPLANE);
      int pos = (int)(id % NPLANE);
      int y = pos / W_, x = pos - y * W_;
      float logit = unkey(selKey[tid]);
      float score = 1.0f / (1.0f + expf(-logit));
      o[0] = (float)x + regs[pos];
      o[1] = (float)y + regs[NPLANE + pos];
      o[2] = wh[pos];
      o[3] = wh[NPLANE + pos];
      o[4] = rot[pos];
      o[5] = score;
      o[6] = (float)c;
    }
  }
}

extern "C" void kernel_launch(void* const* d_in, const int* in_sizes, int n_in,
                              void* d_out, int out_size, void* d_ws, size_t ws_size,
                              hipStream_t stream) {
  const float* hmap = (const float*)d_in[0];
  const float* regs = (const float*)d_in[1];
  const float* wh   = (const float*)d_in[2];
  const float* rot  = (const float*)d_in[3];
  float* out = (float*)d_out;

  unsigned* hist = (unsigned*)d_ws;
  unsigned* meta = hist + NBINS;   // [0]=threshold bin, [1]=count, [3]=fallback flag
  unsigned* vals = meta + 16;
  size_t remaining = ws_size / sizeof(unsigned);
  size_t avail = (remaining > (size_t)(NBINS + 16)) ? (remaining - NBINS - 16) / 2 : 0;
  unsigned cap = (unsigned)((avail < CAPMAX) ? avail : CAPMAX);
  unsigned* ids = vals + cap;

  const int nTiles = C_ * (W_ / TX) * (H_ / TY);   // 80*4*32 = 10240

  zero_ws_kernel<<<(NBINS + 255) / 256, 256, 0, stream>>>(hist, meta);
  nms_hist_kernel<<<nTiles, 256, 0, stream>>>(hmap, hist, meta, vals, ids, cap);
  thresh_kernel<<<1, 32, 0, stream>>>(hist, meta, cap);
  collect_kernel<<<nTiles, 256, 0, stream>>>(hmap, meta, vals, ids, cap);
  finalize_kernel<<<1, 256, 0, stream>>>(regs, wh, rot, meta, vals, ids, out, cap);
}